// PointNetSetAbstraction_83425444757837
// MI455X (gfx1250) — compile-verified
//
#include <hip/hip_runtime.h>

typedef __attribute__((ext_vector_type(2))) float v2f;
typedef __attribute__((ext_vector_type(8))) float v8f;

constexpr int   BATCH    = 8;
constexpr int   NPTS     = 4096;
constexpr int   CIN      = 64;
constexpr int   NPOINT   = 1024;
constexpr int   NSAMPLE  = 32;
constexpr float RADIUS2  = 0.2f * 0.2f;
constexpr float BN_SCALE = 0.99950037460729f;   // 1/sqrt(1 + 1e-3)

// ---------------------------------------------------------------------------
// Kernel 1: farthest point sampling. One block per batch, 1024 threads.
// Point coords live in registers; argmax via wave32 shuffle reduction
// (no barriers) + one 32-entry cross-wave stage => 2 barriers/iteration.
// Winning coords are carried through the reduction so the next pivot
// needs no dependent global load.
// ---------------------------------------------------------------------------
__global__ __launch_bounds__(1024) void fps_kernel(const float* __restrict__ xyz,
                                                   int* __restrict__ fps_idx,
                                                   float* __restrict__ new_xyz) {
  const int b    = blockIdx.x;
  const int tid  = threadIdx.x;
  const int lane = tid & 31;
  const int wv   = tid >> 5;
  const float* px = xyz + (size_t)b * NPTS * 3;

  float x[4], y[4], z[4], dist[4];
#pragma unroll
  for (int i = 0; i < 4; ++i) {
    const int p = i * 1024 + tid;
    x[i] = px[p * 3 + 0];
    y[i] = px[p * 3 + 1];
    z[i] = px[p * 3 + 2];
    dist[i] = 1e10f;
  }

  __shared__ float s_val[32], s_cx[32], s_cy[32], s_cz[32];
  __shared__ int   s_idx[32];
  __shared__ float s_far[3];

  if (tid == 0) {
    s_far[0] = px[0]; s_far[1] = px[1]; s_far[2] = px[2];
    fps_idx[b * NPOINT] = 0;
    float* o = new_xyz + (size_t)b * NPOINT * 3;
    o[0] = px[0]; o[1] = px[1]; o[2] = px[2];
  }
  __syncthreads();

  for (int it = 0; it < NPOINT; ++it) {
    const float fx = s_far[0], fy = s_far[1], fz = s_far[2];

    float best = -1.0f, bx = 0.f, by = 0.f, bz = 0.f;
    int   bi   = tid;
#pragma unroll
    for (int i = 0; i < 4; ++i) {
      const float dx = x[i] - fx, dy = y[i] - fy, dz = z[i] - fz;
      float d = dx * dx + dy * dy + dz * dz;
      d = fminf(dist[i], d);
      dist[i] = d;
      if (d > best) { best = d; bi = i * 1024 + tid; bx = x[i]; by = y[i]; bz = z[i]; }
    }
    // wave32 argmax (prefer lower index on ties)
#pragma unroll
    for (int off = 16; off > 0; off >>= 1) {
      const float ov = __shfl_xor(best, off, 32);
      const int   oi = __shfl_xor(bi,   off, 32);
      const float ox = __shfl_xor(bx,   off, 32);
      const float oy = __shfl_xor(by,   off, 32);
      const float oz = __shfl_xor(bz,   off, 32);
      if (ov > best || (ov == best && oi < bi)) {
        best = ov; bi = oi; bx = ox; by = oy; bz = oz;
      }
    }
    if (lane == 0) { s_val[wv] = best; s_idx[wv] = bi; s_cx[wv] = bx; s_cy[wv] = by; s_cz[wv] = bz; }
    __syncthreads();

    if (wv == 0) {
      float v  = s_val[lane];
      int   ix = s_idx[lane];
      float vx = s_cx[lane], vy = s_cy[lane], vz = s_cz[lane];
#pragma unroll
      for (int off = 16; off > 0; off >>= 1) {
        const float ov = __shfl_xor(v,  off, 32);
        const int   oi = __shfl_xor(ix, off, 32);
        const float ox = __shfl_xor(vx, off, 32);
        const float oy = __shfl_xor(vy, off, 32);
        const float oz = __shfl_xor(vz, off, 32);
        if (ov > v || (ov == v && oi < ix)) { v = ov; ix = oi; vx = ox; vy = oy; vz = oz; }
      }
      if (lane == 0 && it + 1 < NPOINT) {
        fps_idx[b * NPOINT + it + 1] = ix;
        float* o = new_xyz + ((size_t)b * NPOINT + it + 1) * 3;
        o[0] = vx; o[1] = vy; o[2] = vz;
        s_far[0] = vx; s_far[1] = vy; s_far[2] = vz;
      }
    }
    __syncthreads();
  }
}

// ---------------------------------------------------------------------------
// Kernel 2: ball query. One thread per (b,s) query; per-batch xyz is 48KB so
// the scan stays L2-resident. Keeps first NSAMPLE in-range indices in order.
// ---------------------------------------------------------------------------
__global__ __launch_bounds__(256) void ballquery_kernel(const float* __restrict__ xyz,
                                                        const float* __restrict__ new_xyz,
                                                        int* __restrict__ gidx) {
  const int q = blockIdx.x * 256 + threadIdx.x;      // 0 .. 8191 exactly
  const int b = q >> 10;
  const float* px = xyz + (size_t)b * NPTS * 3;
  const float nx = new_xyz[(size_t)q * 3 + 0];
  const float ny = new_xyz[(size_t)q * 3 + 1];
  const float nz = new_xyz[(size_t)q * 3 + 2];
  int* o = gidx + (size_t)q * NSAMPLE;

  int cnt = 0, first = 0;
  for (int i = 0; i < NPTS; ++i) {
    const float dx = px[i * 3 + 0] - nx;
    const float dy = px[i * 3 + 1] - ny;
    const float dz = px[i * 3 + 2] - nz;
    const float d = dx * dx + dy * dy + dz * dz;
    if (d <= RADIUS2) {
      if (cnt == 0) first = i;
      o[cnt++] = i;
      if (cnt == NSAMPLE) break;
    }
  }
  if (cnt == 0) { o[0] = 0; cnt = 1; first = 0; }
  for (; cnt < NSAMPLE; ++cnt) o[cnt] = first;
}

// ---------------------------------------------------------------------------
// Fused MLP: one 16x16 D-tile per wave, K-loop of f32 WMMA (16x16x4).
// pad_col: column of Xin that is zero-padding; W rows above it shift by -1
// (layer 0 keeps points 16B-aligned for async B128 staging).
// ---------------------------------------------------------------------------
__device__ __forceinline__ void mlp_layer(const float* __restrict__ Xin, int ldi,
                                          int Kp, int Kv, int pad_col,
                                          const float* __restrict__ W, int Nc,
                                          const float* __restrict__ bias,
                                          const float* __restrict__ gam,
                                          const float* __restrict__ bet,
                                          float* __restrict__ Xout, int ldo) {
  const int lane = threadIdx.x & 31;
  const int wave = threadIdx.x >> 5;
  const int half = lane >> 4;       // 0 or 1
  const int l16  = lane & 15;
  const int ntiles = Nc >> 4;
  const int total  = 4 * ntiles;    // 4 M-tiles (64 rows) x N-tiles

  for (int t = wave; t < total; t += 8) {
    const int mt   = t / ntiles;
    const int nt   = t - mt * ntiles;
    const int arow = mt * 16 + l16;
    const int ncol = nt * 16 + l16;

    v8f acc = {};
    for (int kk = 0; kk < Kp; kk += 4) {
      const int ka = kk + half * 2;
      const int w0 = ka     - ((ka     > pad_col) ? 1 : 0);
      const int w1 = ka + 1 - ((ka + 1 > pad_col) ? 1 : 0);
      v2f a, bf;
      a.x  = Xin[arow * ldi + ka];
      a.y  = Xin[arow * ldi + ka + 1];
      bf.x = (ka     != pad_col && w0 < Kv) ? W[w0 * Nc + ncol] : 0.0f;
      bf.y = (ka + 1 != pad_col && w1 < Kv) ? W[w1 * Nc + ncol] : 0.0f;
      acc = __builtin_amdgcn_wmma_f32_16x16x4_f32(false, a, false, bf,
                                                  (short)0, acc, false, false);
    }

    const float bi = bias[ncol];
    const float gm = gam[ncol] * BN_SCALE;
    const float be = bet[ncol];
#pragma unroll
    for (int j = 0; j < 8; ++j) {
      const int r = mt * 16 + half * 8 + j;
      float v = acc[j] + bi;
      v = fmaxf(v, 0.0f) * gm + be;
      Xout[r * ldo + ncol] = v;
    }
  }
}

__global__ __launch_bounds__(256) void mlp_fused_kernel(
    const float* __restrict__ xyz, const float* __restrict__ points,
    const float* __restrict__ new_xyz, const int* __restrict__ gidx,
    const float* __restrict__ W0, const float* __restrict__ b0,
    const float* __restrict__ g0, const float* __restrict__ be0,
    const float* __restrict__ W1, const float* __restrict__ b1,
    const float* __restrict__ g1, const float* __restrict__ be1,
    const float* __restrict__ W2, const float* __restrict__ b2,
    const float* __restrict__ g2, const float* __restrict__ be2,
    float* __restrict__ out_feat) {
  __shared__ float bufA[64 * 76];    // X0 (ld 76, points @ col 4), later X2 (ld 68)
  __shared__ float bufB[64 * 132];   // X1 (ld 68), later X3 (ld 132)

  const int tid = threadIdx.x;
  const int blk = blockIdx.x;        // 2 query points per block -> 64 rows

  __builtin_prefetch(W0, 0, 0);
  __builtin_prefetch(W1, 0, 0);
  __builtin_prefetch(W2, 0, 0);

  // Stage X0: cols 0..2 = centered xyz, col 3 = pad, cols 4..67 = points row.
  // The 256B points row is copied with async B128 LDS loads (16B-aligned:
  // stride 76*4=304=16*19, base col 4).
  {
    const int r = tid >> 2;                 // 0..63
    const int c = tid & 3;                  // 64B chunk / pad-col index
    const int q = blk * 2 + (r >> 5);
    const int k = r & 31;
    const int b = q >> 10;
    const int g = gidx[(size_t)q * NSAMPLE + k];

    float v = 0.0f;
    if (c < 3) v = xyz[((size_t)b * NPTS + g) * 3 + c] - new_xyz[(size_t)q * 3 + c];
    bufA[r * 76 + c] = v;

    const unsigned lds_addr = (unsigned)(uintptr_t)(&bufA[r * 76 + 4]) + (unsigned)c * 64u;
    const unsigned goff     = ((unsigned)((b * NPTS + g) * CIN) + (unsigned)c * 16u) * 4u;
#pragma unroll
    for (int i = 0; i < 4; ++i) {
      asm volatile("global_load_async_to_lds_b128 %0, %1, %2"
                   :
                   : "v"(lds_addr + (unsigned)(i * 16)),
                     "v"(goff + (unsigned)(i * 16)),
                     "s"(points)
                   : "memory");
    }
  }
  asm volatile("s_wait_asynccnt 0x0" ::: "memory");
  __syncthreads();

  mlp_layer(bufA, 76, 68, 67,   3, W0,  64, b0, g0, be0, bufB, 68);   // 67 -> 64
  __syncthreads();
  mlp_layer(bufB, 68, 64, 64, 255, W1,  64, b1, g1, be1, bufA, 68);   // 64 -> 64
  __syncthreads();
  mlp_layer(bufA, 68, 64, 64, 255, W2, 128, b2, g2, be2, bufB, 132);  // 64 -> 128
  __syncthreads();

  // Maxpool over the 32 samples of each query point. 256 threads = 2 x 128.
  const int sl = tid >> 7;
  const int ch = tid & 127;
  float m = -3.4e38f;
#pragma unroll 4
  for (int k = 0; k < 32; ++k)
    m = fmaxf(m, bufB[(sl * 32 + k) * 132 + ch]);
  const int q = blk * 2 + sl;
  out_feat[(size_t)q * 128 + ch] = m;
}

// ---------------------------------------------------------------------------
extern "C" void kernel_launch(void* const* d_in, const int* in_sizes, int n_in,
                              void* d_out, int out_size, void* d_ws, size_t ws_size,
                              hipStream_t stream) {
  const float* xyz    = (const float*)d_in[0];
  const float* points = (const float*)d_in[1];
  const float* W0  = (const float*)d_in[2];
  const float* b0  = (const float*)d_in[3];
  const float* g0  = (const float*)d_in[4];
  const float* be0 = (const float*)d_in[5];
  const float* W1  = (const float*)d_in[6];
  const float* b1  = (const float*)d_in[7];
  const float* g1  = (const float*)d_in[8];
  const float* be1 = (const float*)d_in[9];
  const float* W2  = (const float*)d_in[10];
  const float* b2  = (const float*)d_in[11];
  const float* g2  = (const float*)d_in[12];
  const float* be2 = (const float*)d_in[13];

  float* out      = (float*)d_out;
  float* new_xyz  = out;                                   // (8,1024,3)
  float* out_feat = out + (size_t)BATCH * NPOINT * 3;      // (8,1024,128)

  int* fps_idx = (int*)d_ws;                               // 8*1024 ints
  int* gidx    = fps_idx + BATCH * NPOINT;                 // 8192*32 ints

  hipLaunchKernelGGL(fps_kernel, dim3(BATCH), dim3(1024), 0, stream,
                     xyz, fps_idx, new_xyz);
  hipLaunchKernelGGL(ballquery_kernel, dim3((BATCH * NPOINT) / 256), dim3(256), 0, stream,
                     xyz, new_xyz, gidx);
  hipLaunchKernelGGL(mlp_fused_kernel, dim3(BATCH * NPOINT / 2), dim3(256), 0, stream,
                     xyz, points, new_xyz, gidx,
                     W0, b0, g0, be0, W1, b1, g1, be1, W2, b2, g2, be2,
                     out_feat);
}